// APINet_26448408608913
// MI455X (gfx1250) — compile-verified
//
#include <hip/hip_runtime.h>
#include <math.h>

// ---------------- problem constants ----------------
#define NPTS   4096      // N
#define DIM    2048      // D
#define HID    512       // H
#define NCLS   200       // C
#define TWO_N  8192      // 2N
#define KCHUNK 128

typedef float v2f __attribute__((ext_vector_type(2)));
typedef float v4f __attribute__((ext_vector_type(4)));
typedef float v8f __attribute__((ext_vector_type(8)));
typedef int   v4i __attribute__((ext_vector_type(4)));

__device__ __forceinline__ v8f wmma_f32(v2f a, v2f b, v8f c) {
  // V_WMMA_F32_16X16X4_F32 : D = A(16x4) * B(4x16) + C(16x16)
  return __builtin_amdgcn_wmma_f32_16x16x4_f32(
      false, a, false, b, (short)0, c, false, false);
}

__device__ __forceinline__ float sigmoidf(float t) {
  return 1.0f / (1.0f + __expf(-t));
}

// -------- CDNA5 async global->LDS staging (ASYNCcnt path), with fallback ----
#if __has_builtin(__builtin_amdgcn_global_load_async_to_lds_b128)
#define HAVE_ASYNC_LDS 1
#else
#define HAVE_ASYNC_LDS 0
#endif

__device__ __forceinline__ void stage_b128(float* lds, const float* g) {
#if HAVE_ASYNC_LDS
  __builtin_amdgcn_global_load_async_to_lds_b128(
      (__attribute__((address_space(1))) v4i*)(g),
      (__attribute__((address_space(3))) v4i*)(lds), 0, 0);
#else
  *(v4f*)lds = *(const v4f*)g;
#endif
}

__device__ __forceinline__ void stage_wait() {
#if HAVE_ASYNC_LDS
#if __has_builtin(__builtin_amdgcn_s_wait_asynccnt)
  __builtin_amdgcn_s_wait_asynccnt(0);
#else
  asm volatile("s_wait_asynccnt 0x0" ::: "memory");
#endif
#endif
}

// ---------------- kernel 0: 16x16-tiled transpose (for WMMA-B layout) -------
__global__ __launch_bounds__(256) void transpose_kernel(
    const float* __restrict__ in, float* __restrict__ outT, int K, int N) {
  __shared__ float tile[16][17];
  const int kb = blockIdx.x * 16, nb = blockIdx.y * 16;
  const int tx = threadIdx.x & 15, ty = threadIdx.x >> 4;
  int k = kb + ty, n = nb + tx;
  if (k < K && n < N) tile[ty][tx] = in[(size_t)k * N + n];
  __syncthreads();
  int k2 = kb + tx, n2 = nb + ty;
  if (k2 < K && n2 < N) outT[(size_t)n2 * K + k2] = tile[tx][ty];
}

// ---------------- kernel 1: row squared norms ----------------
__global__ __launch_bounds__(256) void sqnorm_kernel(
    const float* __restrict__ feats, float* __restrict__ sq) {
  int wave = (blockIdx.x * blockDim.x + threadIdx.x) >> 5;
  int lane = threadIdx.x & 31;
  if (wave >= NPTS) return;
  const float* row = feats + (size_t)wave * DIM;
  float s = 0.f;
  for (int t = 0; t < DIM; t += 128) {
    v4f v = *(const v4f*)&row[t + lane * 4];
    s = fmaf(v[0], v[0], s); s = fmaf(v[1], v[1], s);
    s = fmaf(v[2], v[2], s); s = fmaf(v[3], v[3], s);
  }
  #pragma unroll
  for (int off = 16; off > 0; off >>= 1) s += __shfl_xor(s, off, 32);
  if (lane == 0) sq[wave] = s;
}

// ---------------- kernel 2: fused Gram (WMMA) + masked argmin ----------------
// Full 16 x DIM A panel staged ONCE per block into dynamic LDS (128 KB) via
// async-b128; the column-tile sweep is then pure wmma with no barriers.
__global__ __launch_bounds__(256) void matchpair_kernel(
    const float* __restrict__ feats, const int* __restrict__ targets,
    const float* __restrict__ sq,
    int* __restrict__ intra_idx, int* __restrict__ inter_idx) {
  extern __shared__ float sA[];             // 16 * DIM floats (dynamic)
  __shared__ float sSq[16];
  __shared__ int   sTgt[16];
  __shared__ unsigned long long red[16][2];

  const int r0   = blockIdx.x * 16;
  const int tid  = threadIdx.x;
  const int wave = tid >> 5, lane = tid & 31;
  const int half = lane >> 4, ln = lane & 15;

  if (tid < 16) { sSq[tid] = sq[r0 + tid]; sTgt[tid] = targets[r0 + tid]; }
  if (tid < 32) red[tid >> 1][tid & 1] = ~0ull;

  // one-shot async stage of the whole A panel (16 rows x DIM)
  for (int i = tid; i < (16 * DIM) / 4; i += 256) {
    int rr = i >> 9;                 // DIM/4 = 512 b128 units per row
    int c4 = (i & 511) << 2;
    stage_b128(&sA[rr * DIM + c4], &feats[(size_t)(r0 + rr) * DIM + c4]);
  }
  stage_wait();
  __syncthreads();

  unsigned long long bestIntra[8], bestInter[8];
  #pragma unroll
  for (int v = 0; v < 8; ++v) { bestIntra[v] = ~0ull; bestInter[v] = ~0ull; }

  const int NCTILE = NPTS / 16;             // 256 column tiles
  for (int g = 0; g < NCTILE; g += 8) {
    const int j0 = (g + wave) * 16;
    const int colLane = j0 + ln;
    const int   tgtCol = targets[colLane];
    const float sqCol  = sq[colLane];
    const float* brow  = feats + (size_t)colLane * DIM;

    v8f c = {};
    for (int kc = 0; kc < DIM; kc += KCHUNK) {
      if (kc + KCHUNK < DIM) __builtin_prefetch(&brow[kc + KCHUNK], 0, 3);
      #pragma unroll 4
      for (int kk = 0; kk < KCHUNK; kk += 4) {
        int kb = kc + kk + 2 * half;
        v2f a = *(const v2f*)&sA[ln * DIM + kb];
        v2f b = *(const v2f*)&brow[kb];
        c = wmma_f32(a, b, c);
      }
    }
    #pragma unroll
    for (int v = 0; v < 8; ++v) {
      int m = v + 8 * half;
      int row = r0 + m;
      float d2 = fmaxf(sSq[m] + sqCol - 2.0f * c[v], 0.0f);
      unsigned long long key =
          ((unsigned long long)__float_as_uint(d2) << 32) | (unsigned)colLane;
      bool same = (sTgt[m] == tgtCol);
      if (same && colLane != row) bestIntra[v] = min(bestIntra[v], key);
      if (!same)                  bestInter[v] = min(bestInter[v], key);
    }
  }
  __syncthreads();
  #pragma unroll
  for (int v = 0; v < 8; ++v) {
    int m = v + 8 * half;
    atomicMin(&red[m][0], bestIntra[v]);
    atomicMin(&red[m][1], bestInter[v]);
  }
  __syncthreads();
  if (tid < 16) {
    intra_idx[r0 + tid] = (int)(unsigned)(red[tid][0] & 0xffffffffull);
    inter_idx[r0 + tid] = (int)(unsigned)(red[tid][1] & 0xffffffffull);
  }
}

// ------------- kernel 3: h = [x||y] @ w1 + b1 (gathered A, w1 transposed) ---
// Double-buffered async staging: next chunk's copies overlap the WMMA loop.
__global__ __launch_bounds__(256) void gemm1_kernel(
    const float* __restrict__ feats, const int* __restrict__ intra_idx,
    const int* __restrict__ inter_idx, const float* __restrict__ w1T,
    const float* __restrict__ b1, float* __restrict__ hout) {
  __shared__ float sA[2][16 * KCHUNK];
  __shared__ int sXrow[16], sYrow[16];

  const int r0   = blockIdx.x * 16;
  const int tid  = threadIdx.x;
  const int wave = tid >> 5, lane = tid & 31;
  const int half = lane >> 4, ln = lane & 15;
  const int KTOT = 2 * DIM;

  if (tid < 16) {
    int r = r0 + tid;
    sXrow[tid] = r & (NPTS - 1);
    sYrow[tid] = (r < NPTS) ? intra_idx[r] : inter_idx[r - NPTS];
  }
  __syncthreads();

  auto stage_chunk = [&](int kc, float* dst) {
    const bool yhalf = (kc >= DIM);
    for (int i = tid; i < (16 * KCHUNK) / 4; i += 256) {
      int rr = i >> 5, c4 = (i & 31) << 2;
      int src = yhalf ? sYrow[rr] : sXrow[rr];
      stage_b128(&dst[rr * KCHUNK + c4],
                 &feats[(size_t)src * DIM + ((kc + c4) & (DIM - 1))]);
    }
  };

  v8f acc[4];
  #pragma unroll
  for (int t = 0; t < 4; ++t) acc[t] = (v8f){};

  stage_chunk(0, sA[0]);
  stage_wait();
  __syncthreads();

  int cur = 0;
  for (int kc = 0; kc < KTOT; kc += KCHUNK) {
    if (kc + KCHUNK < KTOT) stage_chunk(kc + KCHUNK, sA[cur ^ 1]);
    __builtin_prefetch(&w1T[(size_t)(wave * 16 + ln) * KTOT + kc + KCHUNK], 0, 3);
    const float* buf = sA[cur];
    #pragma unroll 4
    for (int kk = 0; kk < KCHUNK; kk += 4) {
      int kb = kk + 2 * half;
      v2f a = *(const v2f*)&buf[ln * KCHUNK + kb];
      int kg = kc + kb;
      #pragma unroll
      for (int t = 0; t < 4; ++t) {
        int col = (wave + 8 * t) * 16 + ln;
        v2f b = *(const v2f*)&w1T[(size_t)col * KTOT + kg];
        acc[t] = wmma_f32(a, b, acc[t]);
      }
    }
    stage_wait();
    __syncthreads();
    cur ^= 1;
  }
  #pragma unroll
  for (int t = 0; t < 4; ++t) {
    int col = (wave + 8 * t) * 16 + ln;
    float bias = b1[col];
    #pragma unroll
    for (int v = 0; v < 8; ++v) {
      int row = r0 + v + 8 * half;
      hout[(size_t)row * HID + col] = acc[t][v] + bias;
    }
  }
}

// ---------------- kernel 4: m = h @ w2 + b2 (w2 transposed) ----------------
__global__ __launch_bounds__(256) void gemm2_kernel(
    const float* __restrict__ hbuf, const float* __restrict__ w2T,
    const float* __restrict__ b2, float* __restrict__ mout) {
  __shared__ float sA[2][16 * KCHUNK];

  const int r0   = blockIdx.x * 16;
  const int ng   = blockIdx.y;          // 4 groups of 32 n-tiles
  const int tid  = threadIdx.x;
  const int wave = tid >> 5, lane = tid & 31;
  const int half = lane >> 4, ln = lane & 15;

  auto stage_chunk = [&](int kc, float* dst) {
    for (int i = tid; i < (16 * KCHUNK) / 4; i += 256) {
      int rr = i >> 5, c4 = (i & 31) << 2;
      stage_b128(&dst[rr * KCHUNK + c4],
                 &hbuf[(size_t)(r0 + rr) * HID + kc + c4]);
    }
  };

  v8f acc[4];
  #pragma unroll
  for (int t = 0; t < 4; ++t) acc[t] = (v8f){};

  stage_chunk(0, sA[0]);
  stage_wait();
  __syncthreads();

  int cur = 0;
  for (int kc = 0; kc < HID; kc += KCHUNK) {
    if (kc + KCHUNK < HID) stage_chunk(kc + KCHUNK, sA[cur ^ 1]);
    const float* buf = sA[cur];
    #pragma unroll 4
    for (int kk = 0; kk < KCHUNK; kk += 4) {
      int kb = kk + 2 * half;
      v2f a = *(const v2f*)&buf[ln * KCHUNK + kb];
      int kg = kc + kb;
      #pragma unroll
      for (int t = 0; t < 4; ++t) {
        int col = (ng * 32 + wave + 8 * t) * 16 + ln;
        v2f b = *(const v2f*)&w2T[(size_t)col * HID + kg];
        acc[t] = wmma_f32(a, b, acc[t]);
      }
    }
    stage_wait();
    __syncthreads();
    cur ^= 1;
  }
  #pragma unroll
  for (int t = 0; t < 4; ++t) {
    int col = (ng * 32 + wave + 8 * t) * 16 + ln;
    float bias = b2[col];
    #pragma unroll
    for (int v = 0; v < 8; ++v) {
      int row = r0 + v + 8 * half;
      mout[(size_t)row * DIM + col] = acc[t][v] + bias;
    }
  }
}

// --------- kernel 5: four gated logit GEMMs in one pass (wfc transposed) ----
__global__ __launch_bounds__(256) void logits_kernel(
    const float* __restrict__ feats, const int* __restrict__ intra_idx,
    const int* __restrict__ inter_idx, const float* __restrict__ mfeat,
    const float* __restrict__ wfcT, const float* __restrict__ bfc,
    float* __restrict__ out) {
  __shared__ float sXS[16 * KCHUNK], sXO[16 * KCHUNK];
  __shared__ float sYS[16 * KCHUNK], sYO[16 * KCHUNK];
  __shared__ int sXrow[16], sYrow[16];

  const int r0   = blockIdx.x * 16;
  const int tid  = threadIdx.x;
  const int wave = tid >> 5, lane = tid & 31;
  const int half = lane >> 4, ln = lane & 15;
  const size_t VOFF = (size_t)TWO_N * NCLS;

  if (tid < 16) {
    int r = r0 + tid;
    sXrow[tid] = r & (NPTS - 1);
    sYrow[tid] = (r < NPTS) ? intra_idx[r] : inter_idx[r - NPTS];
  }

  const int NTILES = (NCLS + 15) / 16;        // 13
  for (int grp = 0; grp < 2; ++grp) {
    const int nt = grp * 8 + wave;
    const bool active = (nt < NTILES);        // wave-uniform
    const int n0 = nt * 16;
    const int col = n0 + ln;
    const bool cok = (col < NCLS);

    v8f acc[4];
    #pragma unroll
    for (int q = 0; q < 4; ++q) acc[q] = (v8f){};

    for (int kc = 0; kc < DIM; kc += KCHUNK) {
      __syncthreads();
      for (int i = tid; i < (16 * KCHUNK) / 4; i += 256) {
        int rr = i >> 5, c4 = (i & 31) << 2;
        int c = kc + c4;
        v4f x = *(const v4f*)&feats[(size_t)sXrow[rr] * DIM + c];
        v4f y = *(const v4f*)&feats[(size_t)sYrow[rr] * DIM + c];
        v4f m = *(const v4f*)&mfeat[(size_t)(r0 + rr) * DIM + c];
        v4f xs, xo, ys, yo;
        #pragma unroll
        for (int e = 0; e < 4; ++e) {
          float gx = sigmoidf(m[e] * x[e]);
          float gy = sigmoidf(m[e] * y[e]);
          xs[e] = fmaf(x[e], gx, x[e]);
          xo[e] = fmaf(x[e], gy, x[e]);
          ys[e] = fmaf(y[e], gy, y[e]);
          yo[e] = fmaf(y[e], gx, y[e]);
        }
        int o = rr * KCHUNK + c4;
        *(v4f*)&sXS[o] = xs; *(v4f*)&sXO[o] = xo;
        *(v4f*)&sYS[o] = ys; *(v4f*)&sYO[o] = yo;
      }
      __syncthreads();
      if (active) {
        #pragma unroll 4
        for (int kk = 0; kk < KCHUNK; kk += 4) {
          int kb = kk + 2 * half;
          int kg = kc + kb;
          v2f b = cok ? *(const v2f*)&wfcT[(size_t)col * DIM + kg] : (v2f){};
          int ai = ln * KCHUNK + kb;
          v2f a0 = *(const v2f*)&sXS[ai];
          v2f a1 = *(const v2f*)&sXO[ai];
          v2f a2 = *(const v2f*)&sYS[ai];
          v2f a3 = *(const v2f*)&sYO[ai];
          acc[0] = wmma_f32(a0, b, acc[0]);
          acc[1] = wmma_f32(a1, b, acc[1]);
          acc[2] = wmma_f32(a2, b, acc[2]);
          acc[3] = wmma_f32(a3, b, acc[3]);
        }
      }
    }
    if (active && cok) {
      float bias = bfc[col];
      #pragma unroll
      for (int q = 0; q < 4; ++q) {
        #pragma unroll
        for (int v = 0; v < 8; ++v) {
          int row = r0 + v + 8 * half;
          out[(size_t)q * VOFF + (size_t)row * NCLS + col] =
              sigmoidf(acc[q][v] + bias);
        }
      }
    }
  }
}

// ---------------- kernel 6: labels ----------------
__global__ __launch_bounds__(256) void labels_kernel(
    const int* __restrict__ targets, const int* __restrict__ inter_idx,
    float* __restrict__ out) {
  int i = blockIdx.x * blockDim.x + threadIdx.x;
  if (i >= TWO_N) return;
  const size_t base = (size_t)4 * TWO_N * NCLS;
  out[base + i] = (float)targets[i & (NPTS - 1)];
  int l2 = (i < NPTS) ? targets[i] : targets[inter_idx[i - NPTS]];
  out[base + TWO_N + i] = (float)l2;
}

// ---------------- launch ----------------
extern "C" void kernel_launch(void* const* d_in, const int* in_sizes, int n_in,
                              void* d_out, int out_size, void* d_ws, size_t ws_size,
                              hipStream_t stream) {
  const float* feats   = (const float*)d_in[0];
  const int*   targets = (const int*)d_in[1];
  const float* w1      = (const float*)d_in[2];
  const float* b1      = (const float*)d_in[3];
  const float* w2      = (const float*)d_in[4];
  const float* b2      = (const float*)d_in[5];
  const float* wfc     = (const float*)d_in[6];
  const float* bfc     = (const float*)d_in[7];
  float* out = (float*)d_out;

  char* ws = (char*)d_ws;
  float* sq        = (float*)(ws);                                   // 16 KB
  int*   intra_idx = (int*)(ws + 16384);                             // 16 KB
  int*   inter_idx = (int*)(ws + 32768);                             // 16 KB
  float* hbuf      = (float*)(ws + 65536);                           // 16 MB
  float* mbuf      = (float*)(ws + 65536 + (size_t)TWO_N * HID * 4); // 64 MB
  size_t offT = 65536 + (size_t)TWO_N * HID * 4 + (size_t)TWO_N * DIM * 4;
  float* w1T  = (float*)(ws + offT);                                 // 8 MB
  float* w2T  = (float*)(ws + offT + (size_t)2 * DIM * HID * 4);     // 4 MB
  float* wfcT = (float*)(ws + offT + (size_t)2 * DIM * HID * 4
                             + (size_t)HID * DIM * 4);               // 1.6 MB

  sqnorm_kernel<<<NPTS * 32 / 256, 256, 0, stream>>>(feats, sq);
  transpose_kernel<<<dim3(2 * DIM / 16, HID / 16), 256, 0, stream>>>(
      w1, w1T, 2 * DIM, HID);
  transpose_kernel<<<dim3(HID / 16, DIM / 16), 256, 0, stream>>>(
      w2, w2T, HID, DIM);
  transpose_kernel<<<dim3(DIM / 16, (NCLS + 15) / 16), 256, 0, stream>>>(
      wfc, wfcT, DIM, NCLS);
  matchpair_kernel<<<NPTS / 16, 256, 16 * DIM * sizeof(float), stream>>>(
      feats, targets, sq, intra_idx, inter_idx);
  gemm1_kernel<<<TWO_N / 16, 256, 0, stream>>>(feats, intra_idx, inter_idx,
                                               w1T, b1, hbuf);
  gemm2_kernel<<<dim3(TWO_N / 16, 4), 256, 0, stream>>>(hbuf, w2T, b2, mbuf);
  logits_kernel<<<TWO_N / 16, 256, 0, stream>>>(feats, intra_idx, inter_idx,
                                                mbuf, wfcT, bfc, out);
  labels_kernel<<<TWO_N / 256, 256, 0, stream>>>(targets, inter_idx, out);
}